// CQVAELoss_28089086116241
// MI455X (gfx1250) — compile-verified
//
#include <hip/hip_runtime.h>
#include <math.h>

// ---------------------------------------------------------------------------
// CQVAE loss on MI455X (gfx1250, wave32).
// Bandwidth-bound streaming reduction: ~173 MB in, 4 B out -> HBM roofline
// ~7.5us @ 23.3 TB/s. Non-temporal b128 loads, deterministic two-pass
// reduction, WMMA (v_wmma_f32_16x16x4_f32) used for the exact in-wave sum.
// ---------------------------------------------------------------------------

typedef __attribute__((ext_vector_type(2))) float v2f;
typedef __attribute__((ext_vector_type(4))) float v4f;
typedef __attribute__((ext_vector_type(8))) float v8f;

#define BB     128
#define SS     256
#define SG     128
#define DD     1024
#define PP     118
#define NCOMP  6
#define NBLOCKS 512
#define NTHREADS 256
#define ALPHA_W 10.0f
#define EPS_K   1e-20f

// Exact f32 sum across the 32 lanes of a wave using one WMMA:
//   A (16x4 f32): VGPR0 = lane partial  -> A[m][0] (lanes 0-15), A[m][2] (16-31)
//                 VGPR1 = 0             -> A[m][1] = A[m][3] = 0
//   B (4x16 f32): all ones
//   D[m][n] = partial[m] + partial[m+16]  for every n
// Lane n (n<16) holds D[0..7][n] in its 8 D-VGPRs, lane n+16 holds D[8..15][n],
// so sum(own 8) + shfl_xor(16) = full 32-lane sum, identical in all lanes.
// Requires EXEC all-ones: only called with every lane of the wave active.
__device__ __forceinline__ float wave_reduce_wmma(float v) {
    v2f a; a.x = v;    a.y = 0.0f;
    v2f b; b.x = 1.0f; b.y = 1.0f;
    v8f c = {};
    v8f d = __builtin_amdgcn_wmma_f32_16x16x4_f32(
        /*neg_a=*/false, a, /*neg_b=*/false, b,
        /*c_mod=*/(short)0, c, /*reuse_a=*/false, /*reuse_b=*/false);
    float s = d[0] + d[1] + d[2] + d[3] + d[4] + d[5] + d[6] + d[7];
    s += __shfl_xor(s, 16, 32);
    return s;
}

__device__ __forceinline__ bool is_mark(int p) {
    return (p == 0) | (p == 29) | (p == 88) | (p == 117);
}

// Pass 1: every thread grid-strides all four streams, accumulating 6 partial
// sums in registers; block reduces (WMMA per wave + tiny LDS fold) and writes
// 6 floats per block into the workspace.
__global__ void cqvae_partials(const float* __restrict__ zs,
                               const float* __restrict__ rzs,
                               const float* __restrict__ pts,
                               const float* __restrict__ best,
                               const float* __restrict__ qy,
                               const float* __restrict__ gts,
                               const float* __restrict__ best_gt,
                               const int*   __restrict__ mapping,
                               const int*   __restrict__ vdims,
                               float* __restrict__ partials) {
    const int tid = threadIdx.x;
    const int gid = blockIdx.x * blockDim.x + tid;
    const int gsz = gridDim.x * blockDim.x;

    float acc_kld = 0.f, acc_ae = 0.f;
    float acc_ba = 0.f, acc_bm = 0.f;        // bias (gathered pts vs gts)
    float acc_ca = 0.f, acc_cm = 0.f;        // best vs best_gt

    // ---- ae_loss: sum (rzs[b, map[b,i], d] - zs[b, i, d])^2  (float4) ----
    {
        const int rowq = DD / 4;                       // 256 float4 per row
        const int n4 = BB * SG * rowq;
        const v4f* rz4 = (const v4f*)rzs;
        const v4f* z4  = (const v4f*)zs;
        for (int e = gid; e < n4; e += gsz) {
            int d4 = e & (rowq - 1);
            int bi = e >> 8;                           // / rowq
            int i  = bi & (SG - 1);
            int b  = bi >> 7;                          // / SG
            int m  = mapping[bi];
            v4f r = __builtin_nontemporal_load(rz4 + ((b * SS + m) * rowq + d4));
            v4f z = __builtin_nontemporal_load(z4  + ((b * SS + i) * rowq + d4));
            v4f df = r - z;
            acc_ae += df.x * df.x + df.y * df.y + df.z * df.z + df.w * df.w;
        }
    }

    // ---- kld_loss: sum qy * (log(qy + eps) - log(1/V))  (float4) ----
    {
        const int V = vdims[0];
        const float logU = -logf((float)V);            // log(1/V)
        const int n4 = (BB * SS * V) / 4;
        const v4f* q4 = (const v4f*)qy;
        for (int e = gid; e < n4; e += gsz) {
            v4f q = __builtin_nontemporal_load(q4 + e);
            acc_kld += q.x * (logf(q.x + EPS_K) - logU)
                     + q.y * (logf(q.y + EPS_K) - logU)
                     + q.z * (logf(q.z + EPS_K) - logU)
                     + q.w * (logf(q.w + EPS_K) - logU);
        }
    }

    // ---- bias_loss: gathered pts vs gts, one (x,y) point per iter ----
    {
        const int n = BB * SG * PP;
        const v2f* pt2 = (const v2f*)pts;
        const v2f* gt2 = (const v2f*)gts;
        for (int e = gid; e < n; e += gsz) {
            int p  = e % PP;
            int bi = e / PP;
            int i  = bi & (SG - 1);
            int b  = bi >> 7;
            int m  = mapping[bi];
            v2f pg = __builtin_nontemporal_load(pt2 + ((b * SS + m) * PP + p));
            v2f g  = __builtin_nontemporal_load(gt2 + (bi * PP + p));
            v2f df = pg - g;
            float sq = df.x * df.x + df.y * df.y;
            acc_ba += sq;
            if (is_mark(p)) acc_bm += sq;
        }
    }

    // ---- best_mse: best vs best_gt (tiny) ----
    {
        const int n = BB * PP;
        const v2f* b2 = (const v2f*)best;
        const v2f* g2 = (const v2f*)best_gt;
        for (int e = gid; e < n; e += gsz) {
            int p = e % PP;
            v2f df = b2[e] - g2[e];
            float sq = df.x * df.x + df.y * df.y;
            acc_ca += sq;
            if (is_mark(p)) acc_cm += sq;
        }
    }

    // ---- block reduction: WMMA per wave, LDS fold across the 8 waves ----
    float comp[NCOMP] = {acc_kld, acc_ae, acc_ba, acc_bm, acc_ca, acc_cm};
    __shared__ float lds[NCOMP][NTHREADS / 32];
    const int wave = tid >> 5, lane = tid & 31;
    #pragma unroll
    for (int c = 0; c < NCOMP; ++c) {
        float s = wave_reduce_wmma(comp[c]);
        if (lane == 0) lds[c][wave] = s;
    }
    __syncthreads();
    if (tid == 0) {
        #pragma unroll
        for (int c = 0; c < NCOMP; ++c) {
            float s = 0.f;
            #pragma unroll
            for (int w = 0; w < NTHREADS / 32; ++w) s += lds[c][w];
            partials[blockIdx.x * NCOMP + c] = s;
        }
    }
}

// Pass 2: single block folds NBLOCKS x 6 partials and applies the weights.
__global__ void cqvae_finalize(const float* __restrict__ partials,
                               const int*   __restrict__ vdims,
                               float* __restrict__ out) {
    const int tid = threadIdx.x;
    float comp[NCOMP] = {0.f, 0.f, 0.f, 0.f, 0.f, 0.f};
    for (int i = tid; i < NBLOCKS; i += blockDim.x) {
        #pragma unroll
        for (int c = 0; c < NCOMP; ++c) comp[c] += partials[i * NCOMP + c];
    }
    __shared__ float lds[NCOMP][NTHREADS / 32];
    const int wave = tid >> 5, lane = tid & 31;
    #pragma unroll
    for (int c = 0; c < NCOMP; ++c) {
        float s = wave_reduce_wmma(comp[c]);
        if (lane == 0) lds[c][wave] = s;
    }
    __syncthreads();
    if (tid == 0) {
        float sums[NCOMP];
        #pragma unroll
        for (int c = 0; c < NCOMP; ++c) {
            float s = 0.f;
            #pragma unroll
            for (int w = 0; w < NTHREADS / 32; ++w) s += lds[c][w];
            sums[c] = s;
        }
        (void)vdims;
        const float kld  = sums[0] / (float)(BB * SS);
        const float ae   = sums[1] / (float)((long)BB * SG * DD);
        const float bias = sums[2] / (float)(BB * SG * PP * 2)
                         + ALPHA_W * sums[3] / (float)(BB * SG * 4 * 2);
        const float bst  = sums[4] / (float)(BB * PP * 2)
                         + ALPHA_W * sums[5] / (float)(BB * 4 * 2);
        // loss = BETA*kld + GAMMA*ae + best_mse + bias   (BETA = GAMMA = 1)
        out[0] = kld + ae + bst + bias;
    }
}

extern "C" void kernel_launch(void* const* d_in, const int* in_sizes, int n_in,
                              void* d_out, int out_size, void* d_ws, size_t ws_size,
                              hipStream_t stream) {
    const float* zs      = (const float*)d_in[0];
    const float* rzs     = (const float*)d_in[1];
    const float* pts     = (const float*)d_in[2];
    const float* best    = (const float*)d_in[3];
    const float* qy      = (const float*)d_in[4];
    const float* gts     = (const float*)d_in[5];
    const float* best_gt = (const float*)d_in[6];
    const int*   mapping = (const int*)d_in[7];
    const int*   vdims   = (const int*)d_in[8];
    float* partials = (float*)d_ws;   // NBLOCKS * 6 floats, fully overwritten

    cqvae_partials<<<NBLOCKS, NTHREADS, 0, stream>>>(
        zs, rzs, pts, best, qy, gts, best_gt, mapping, vdims, partials);
    cqvae_finalize<<<1, NTHREADS, 0, stream>>>(partials, vdims, (float*)d_out);
}